// GaussianKernel_62792421868327
// MI455X (gfx1250) — compile-verified
//
#include <hip/hip_runtime.h>
#include <hip/hip_bf16.h>

typedef __attribute__((ext_vector_type(2))) float v2f;
typedef __attribute__((ext_vector_type(8))) float v8f;

#define M_TOTAL 1024   // 4 * 256 test points
#define N_TOTAL 4096   // train points
#define K_DIM   64
#define INV2VAR 312.5f                       // 1 / (2 * 0.04^2)
#define OUTSCALE (9.973557010039583f / 4096.0f)  // coef / N, coef = 1/sqrt(2*pi*var)

// Kernel 1: squared norms of every test/train row + zero-init the atomic output.
__global__ __launch_bounds__(256) void kde_norms_kernel(
    const float* __restrict__ X, const float* __restrict__ Y,
    float* __restrict__ xn2, float* __restrict__ yn2, float* __restrict__ out)
{
    int idx = blockIdx.x * blockDim.x + threadIdx.x;
    if (idx < N_TOTAL) {
        const float4* p = (const float4*)(Y + (size_t)idx * K_DIM);
        float s = 0.f;
#pragma unroll
        for (int i = 0; i < K_DIM / 4; ++i) {
            float4 v = p[i];
            s += v.x * v.x + v.y * v.y + v.z * v.z + v.w * v.w;
        }
        yn2[idx] = s;
    }
    if (idx < M_TOTAL) {
        const float4* p = (const float4*)(X + (size_t)idx * K_DIM);
        float s = 0.f;
#pragma unroll
        for (int i = 0; i < K_DIM / 4; ++i) {
            float4 v = p[i];
            s += v.x * v.x + v.y * v.y + v.z * v.z + v.w * v.w;
        }
        xn2[idx] = s;
        out[idx] = 0.f;   // zero before atomic accumulation (stream-ordered)
    }
}

// Kernel 2: one wave per (16-row M tile, 64-col N chunk).
// f32 WMMA 16x16x4 computes the exact-precision Gram tile; then exp + column
// reduction + atomicAdd of per-row partial sums.
__global__ __launch_bounds__(32) void kde_wmma_kernel(
    const float* __restrict__ X, const float* __restrict__ Y,
    const float* __restrict__ xn2, const float* __restrict__ yn2,
    float* __restrict__ out)
{
    const int lane = threadIdx.x;       // 0..31 (wave32)
    const int row  = lane & 15;         // matrix row (A) / col (B,C) index
    const int hi   = lane >> 4;         // K-half selector for A/B fragments
    const int mbase  = blockIdx.x * 16;
    const int nbase0 = blockIdx.y * 64; // 4 N-tiles of 16

    // A fragments: 16x4 f32 layout — lanes 0-15: M=lane, K={kb,kb+1};
    //              lanes 16-31: M=lane-16, K={kb+2,kb+3}.
    v2f a[16];
    const float* xrow = X + (size_t)(mbase + row) * K_DIM + 2 * hi;
#pragma unroll
    for (int kk = 0; kk < 16; ++kk)
        a[kk] = *(const v2f*)(xrow + 4 * kk);

    // xn2 for the 8 rows this lane's C fragment covers (vgpr r -> M = r + 8*hi)
    float xr[8];
#pragma unroll
    for (int r = 0; r < 8; ++r)
        xr[r] = xn2[mbase + 8 * hi + r];

    float sum[8];
#pragma unroll
    for (int r = 0; r < 8; ++r) sum[r] = 0.f;

    for (int t = 0; t < 4; ++t) {
        const int nbase = nbase0 + t * 16;
        // B fragments: 4x16 f32, N across lanes, same K-half split as A.
        const float* yrow = Y + (size_t)(nbase + row) * K_DIM + 2 * hi;
        v8f c = {};
#pragma unroll
        for (int kk = 0; kk < 16; ++kk) {
            v2f b = *(const v2f*)(yrow + 4 * kk);
            // 8 args: (neg_a, A, neg_b, B, c_mod, C, reuse_a, reuse_b)
            c = __builtin_amdgcn_wmma_f32_16x16x4_f32(
                    false, a[kk], false, b, (short)0, c, false, false);
        }

        const float yv = yn2[nbase + row];  // this lane's column norm
#pragma unroll
        for (int r = 0; r < 8; ++r) {
            // exponent = (2*dot - ||x||^2 - ||y||^2) / (2*var)   (<= 0)
            float e = (2.0f * c[r] - xr[r] - yv) * INV2VAR;
            sum[r] += __expf(e);   // v_exp_f32 fast path
        }
    }

    // Reduce over the 16 columns: xor-shuffles with masks 8..1 stay within
    // each 16-lane half, so both halves reduce their own row set in parallel.
#pragma unroll
    for (int r = 0; r < 8; ++r) {
#pragma unroll
        for (int m = 8; m >= 1; m >>= 1)
            sum[r] += __shfl_xor(sum[r], m, 32);
    }

    if (row == 0) {
#pragma unroll
        for (int r = 0; r < 8; ++r)
            atomicAdd(out + mbase + 8 * hi + r, sum[r] * OUTSCALE);
    }
}

extern "C" void kernel_launch(void* const* d_in, const int* in_sizes, int n_in,
                              void* d_out, int out_size, void* d_ws, size_t ws_size,
                              hipStream_t stream) {
    const float* X = (const float*)d_in[0];  // test_Xs  [4,256,64] -> [1024,64]
    const float* Y = (const float*)d_in[1];  // train_Xs [4096,64]
    float* out = (float*)d_out;              // [1024]
    float* xn2 = (float*)d_ws;               // 1024 floats
    float* yn2 = xn2 + M_TOTAL;              // 4096 floats

    kde_norms_kernel<<<dim3((N_TOTAL + 255) / 256), dim3(256), 0, stream>>>(
        X, Y, xn2, yn2, out);
    kde_wmma_kernel<<<dim3(M_TOTAL / 16, N_TOTAL / 64), dim3(32), 0, stream>>>(
        X, Y, xn2, yn2, out);
}